// SAE_loss_CL_26534307955029
// MI455X (gfx1250) — compile-verified
//
#include <hip/hip_runtime.h>
#include <math.h>

#define NROWS 4096
#define DCOLS 1024
#define KBINS 128
#define STEPW 0.175f

typedef float v2f __attribute__((ext_vector_type(2)));
typedef float v8f __attribute__((ext_vector_type(8)));

#if defined(__gfx1250__)
#if __has_builtin(__builtin_amdgcn_wmma_f32_16x16x4_f32)
#define USE_WMMA 1
#endif
#endif
#ifndef USE_WMMA
#define USE_WMMA 0
#endif

// ---- workspace layout (floats) ----
// [0..2]  ordered-uint min accumulators for data1/2/3
// [3]     sum of squared diffs (mse)
// [4..10] row-entropy sums: h1,h2,h3,h13,h23,h12,h123
// [16..]  per-(block k, row b) stats: 6 floats {m,S,U,T,mo,SO}, k-major

__device__ __forceinline__ unsigned fenc(float f) {
  unsigned u = __float_as_uint(f);
  return (u & 0x80000000u) ? ~u : (u | 0x80000000u);
}
__device__ __forceinline__ float fdec(unsigned u) {
  u = (u & 0x80000000u) ? (u & 0x7FFFFFFFu) : ~u;
  return __uint_as_float(u);
}
__device__ __forceinline__ int binIdx(float x, float lower) {
  int i = (int)(ceilf((x - lower) / STEPW) - 1.0f);
  return i < 0 ? 0 : (i > (KBINS - 1) ? (KBINS - 1) : i);
}

// ---------------- init ----------------
__global__ void init_kernel(float* ws) {
  int t = threadIdx.x;
  if (t < 3) ((unsigned*)ws)[t] = 0xFFFFFFFFu;  // min identity (ordered-uint)
  else if (t < 11) ws[t] = 0.0f;
}

// ---------------- stats: per-row {m,S,U,T,mo,SO} + global min ----------------
__global__ __launch_bounds__(256) void stats_kernel(
    const float* __restrict__ d1, const float* __restrict__ d2, const float* __restrict__ d3,
    const float* __restrict__ o1, const float* __restrict__ o2, const float* __restrict__ o3,
    float* __restrict__ ws) {
  const int k = blockIdx.y;
  const float* __restrict__ D = (k == 0) ? d1 : ((k == 1) ? d2 : d3);
  const float* __restrict__ O = (k == 0) ? o1 : ((k == 1) ? o2 : o3);
  const int tile = blockIdx.x;
  const int t = threadIdx.x;

  __shared__ float sD[16 * 132];  // stride 132 -> bank = (4*row + col) % 64, conflict-free
  __shared__ float sO[16 * 132];
  __shared__ float sM[16], sMO[16];
  __shared__ float acc[4][16];
  __shared__ unsigned sMinU;

  if (t == 0) sMinU = 0xFFFFFFFFu;
  if (t < 64) acc[t >> 4][t & 15] = 0.0f;

  const int r = t >> 4, li = t & 15;
  const size_t rowbase = (size_t)tile * 16;

  // phase 1: streaming row max/min (coalesced, no LDS)
  float mx = -INFINITY, mxo = -INFINITY, mn = INFINITY;
  {
    const float* Dr = D + (rowbase + r) * (size_t)DCOLS;
    const float* Or = O + (rowbase + r) * (size_t)DCOLS;
    for (int c = li; c < DCOLS; c += 16) {
      float dv = Dr[c], ov = Or[c];
      mx = fmaxf(mx, dv); mn = fminf(mn, dv); mxo = fmaxf(mxo, ov);
    }
  }
  for (int s = 8; s >= 1; s >>= 1) {
    mx  = fmaxf(mx,  __shfl_xor(mx,  s, 16));
    mxo = fmaxf(mxo, __shfl_xor(mxo, s, 16));
    mn  = fminf(mn,  __shfl_xor(mn,  s, 16));
  }
  if (li == 0) { sM[r] = mx; sMO[r] = mxo; }
  __syncthreads();
  if (li == 0) atomicMin(&sMinU, fenc(mn));
  __syncthreads();
  if (t == 0) atomicMin((unsigned*)ws + k, sMinU);

  // phase 2: chunked LDS staging + WMMA row-sum accumulation (B = ones)
  const int lane = t & 31;
  const int wrow = lane & 15;        // A-matrix M = lane % 16
  const int kh = (lane >> 4) << 1;   // K offset: lanes 0-15 -> K 0,1 ; 16-31 -> K 2,3
  const int wave = t >> 5;
  const float m = sM[wrow], mo = sMO[wrow];
  v8f cS = {0, 0, 0, 0, 0, 0, 0, 0};
  v8f cU = cS, cT = cS, cF = cS;
  const v2f bO = {1.0f, 1.0f};

  for (int c0 = 0; c0 < DCOLS; c0 += 128) {
    __syncthreads();
    for (int i = 0; i < 2; i++) {  // stage 16 rows x 128 cols of D and O
      int f4 = t + i * 256;        // 0..511 float4s
      int rr = f4 >> 5;
      int c4 = f4 & 31;
      const float4 vd = *(const float4*)(D + (rowbase + rr) * (size_t)DCOLS + c0 + c4 * 4);
      const float4 vo = *(const float4*)(O + (rowbase + rr) * (size_t)DCOLS + c0 + c4 * 4);
      *(float4*)&sD[rr * 132 + c4 * 4] = vd;
      *(float4*)&sO[rr * 132 + c4 * 4] = vo;
    }
    __syncthreads();
    const int cw = wave * 16;      // each of 8 waves owns 16 columns
    for (int cc = 0; cc < 16; cc += 4) {
      const int c = cw + cc + kh;
      float dv0 = sD[wrow * 132 + c],     dv1 = sD[wrow * 132 + c + 1];
      float ov0 = sO[wrow * 132 + c],     ov1 = sO[wrow * 132 + c + 1];
      float e0 = __expf(dv0 - m),  e1 = __expf(dv1 - m);
      float f0 = __expf(ov0 - mo), f1 = __expf(ov1 - mo);
      v2f aS = {e0, e1};
      v2f aU = {e0 * dv0, e1 * dv1};
      v2f aT = {e0 * ov0, e1 * ov1};
      v2f aF = {f0, f1};
#if USE_WMMA
      cS = __builtin_amdgcn_wmma_f32_16x16x4_f32(false, aS, false, bO, (short)0, cS, false, false);
      cU = __builtin_amdgcn_wmma_f32_16x16x4_f32(false, aU, false, bO, (short)0, cU, false, false);
      cT = __builtin_amdgcn_wmma_f32_16x16x4_f32(false, aT, false, bO, (short)0, cT, false, false);
      cF = __builtin_amdgcn_wmma_f32_16x16x4_f32(false, aF, false, bO, (short)0, cF, false, false);
#else
      (void)bO;
      cS[0] += aS[0] + aS[1]; cU[0] += aU[0] + aU[1];
      cT[0] += aT[0] + aT[1]; cF[0] += aF[0] + aF[1];
#endif
    }
  }

#if USE_WMMA
  // C layout: lane0 holds rows 0..7 in elems 0..7 (N=0), lane16 holds rows 8..15
  if (lane == 0 || lane == 16) {
    int ro = (lane == 16) ? 8 : 0;
    for (int i2 = 0; i2 < 8; i2++) {
      atomicAdd(&acc[0][ro + i2], cS[i2]);
      atomicAdd(&acc[1][ro + i2], cU[i2]);
      atomicAdd(&acc[2][ro + i2], cT[i2]);
      atomicAdd(&acc[3][ro + i2], cF[i2]);
    }
  }
#else
  {
    float pS = cS[0], pU = cU[0], pT = cT[0], pF = cF[0];
    pS += __shfl_xor(pS, 16, 32); pU += __shfl_xor(pU, 16, 32);
    pT += __shfl_xor(pT, 16, 32); pF += __shfl_xor(pF, 16, 32);
    if (lane < 16) {
      atomicAdd(&acc[0][lane], pS); atomicAdd(&acc[1][lane], pU);
      atomicAdd(&acc[2][lane], pT); atomicAdd(&acc[3][lane], pF);
    }
  }
#endif
  __syncthreads();
  if (t < 16) {
    float* st = ws + 16 + ((size_t)k * NROWS + rowbase + t) * 6;
    st[0] = sM[t];      // m
    st[1] = acc[0][t];  // S
    st[2] = acc[1][t];  // U
    st[3] = acc[2][t];  // T
    st[4] = sMO[t];     // mo
    st[5] = acc[3][t];  // SO
  }
}

// ---------------- entropy helpers ----------------
__device__ __forceinline__ void bitonic_sort(unsigned* key, int t) {
  for (unsigned ksz = 2; ksz <= 1024; ksz <<= 1) {
    for (unsigned j = ksz >> 1; j > 0; j >>= 1) {
      __syncthreads();
      for (int e = t; e < 1024; e += 256) {
        int p = e ^ (int)j;
        if (p > e) {
          unsigned a = key[e], c = key[p];
          bool up = ((e & (int)ksz) == 0);
          if (up ? (a > c) : (a < c)) { key[e] = c; key[p] = a; }
        }
      }
    }
  }
  __syncthreads();
}

__device__ __forceinline__ void run_entropy(const unsigned* key, int t, unsigned shift, float* dst) {
  float loc = 0.0f;
  for (int p = t; p < 1024; p += 256) {
    unsigned kp = key[p] >> shift;
    if (p == 0 || (key[p - 1] >> shift) != kp) {
      int q = p + 1;
      while (q < 1024 && (key[q] >> shift) == kp) q++;
      float L = (float)(q - p);
      loc += L * __logf(L);
    }
  }
  if (loc != 0.0f) atomicAdd(dst, loc);
}

// ---------------- per-row discrete entropies ----------------
__global__ __launch_bounds__(256) void entropy_kernel(
    const float* __restrict__ d1, const float* __restrict__ d2, const float* __restrict__ d3,
    float* __restrict__ ws) {
  __shared__ unsigned pc[1024];
  __shared__ unsigned key[1024];
  __shared__ unsigned hist[3][KBINS];
  __shared__ float facc[7];  // sum L*lnL for: 0:i1 1:i2 2:i3 3:(i1,i2) 4:(i1,i2,i3) 5:(i2,i3) 6:(i1,i3)
  const int t = threadIdx.x;
  const size_t b = blockIdx.x;

  for (int i = t; i < 3 * KBINS; i += 256) ((unsigned*)hist)[i] = 0u;
  if (t < 7) facc[t] = 0.0f;
  const float lo1 = floorf(fdec(((const unsigned*)ws)[0]));
  const float lo2 = floorf(fdec(((const unsigned*)ws)[1]));
  const float lo3 = floorf(fdec(((const unsigned*)ws)[2]));
  __syncthreads();

  for (int j = t; j < 1024; j += 256) {
    int i1 = binIdx(d1[b * DCOLS + j], lo1);
    int i2 = binIdx(d2[b * DCOLS + j], lo2);
    int i3 = binIdx(d3[b * DCOLS + j], lo3);
    pc[j] = ((unsigned)i1 << 14) | ((unsigned)i2 << 7) | (unsigned)i3;
    atomicAdd(&hist[0][i1], 1u);
    atomicAdd(&hist[1][i2], 1u);
    atomicAdd(&hist[2][i3], 1u);
  }
  __syncthreads();
  if (t < KBINS) {
    for (int k = 0; k < 3; k++) {
      unsigned c = hist[k][t];
      if (c > 0u) atomicAdd(&facc[k], (float)c * __logf((float)c));
    }
  }

  // sort by (i1,i2,i3): runs give {123} and (>>7) gives {12}
  for (int j = t; j < 1024; j += 256) key[j] = pc[j];
  bitonic_sort(key, t);
  run_entropy(key, t, 0u, &facc[4]);
  run_entropy(key, t, 7u, &facc[3]);
  __syncthreads();
  // sort by (i2,i3): runs give {23}
  for (int j = t; j < 1024; j += 256) key[j] = pc[j] & 0x3FFFu;
  bitonic_sort(key, t);
  run_entropy(key, t, 0u, &facc[5]);
  __syncthreads();
  // sort by (i1,i3): runs give {13}
  for (int j = t; j < 1024; j += 256) key[j] = ((pc[j] >> 14) << 7) | (pc[j] & 0x7Fu);
  bitonic_sort(key, t);
  run_entropy(key, t, 0u, &facc[6]);
  __syncthreads();

  if (t == 0) {
    const float LN1024 = 6.931471805599453f;
    const float inv = 1.0f / 1024.0f;
    atomicAdd(&ws[4],  LN1024 - facc[0] * inv);  // h1
    atomicAdd(&ws[5],  LN1024 - facc[1] * inv);  // h2
    atomicAdd(&ws[6],  LN1024 - facc[2] * inv);  // h3
    atomicAdd(&ws[7],  LN1024 - facc[6] * inv);  // h13
    atomicAdd(&ws[8],  LN1024 - facc[5] * inv);  // h23
    atomicAdd(&ws[9],  LN1024 - facc[3] * inv);  // h12
    atomicAdd(&ws[10], LN1024 - facc[4] * inv);  // h123
  }
}

// ---------------- mse ----------------
__global__ __launch_bounds__(256) void mse_kernel(const float* __restrict__ d,
                                                  const float* __restrict__ o,
                                                  float* __restrict__ ws) {
  const size_t N4 = (size_t)NROWS * 3072 / 4;
  const float4* dp = (const float4*)d;
  const float4* op = (const float4*)o;
  const size_t stride = (size_t)gridDim.x * 256;
  float s = 0.0f;
  for (size_t i = (size_t)blockIdx.x * 256 + threadIdx.x; i < N4; i += stride) {
    __builtin_prefetch(&dp[i + stride], 0, 1);  // global_prefetch_b8
    __builtin_prefetch(&op[i + stride], 0, 1);
    float4 a = dp[i], bv = op[i];
    float x = a.x - bv.x, y = a.y - bv.y, z = a.z - bv.z, w = a.w - bv.w;
    s = fmaf(x, x, s); s = fmaf(y, y, s); s = fmaf(z, z, s); s = fmaf(w, w, s);
  }
  for (int sh = 16; sh >= 1; sh >>= 1) s += __shfl_xor(s, sh, 32);
  if ((threadIdx.x & 31) == 0) atomicAdd(&ws[3], s);
}

// ---------------- final combine ----------------
__global__ __launch_bounds__(256) void final_kernel(const float* __restrict__ ws,
                                                    float* __restrict__ out) {
  __shared__ float red[14];
  const int t = threadIdx.x;
  if (t < 14) red[t] = 0.0f;
  __syncthreads();
  const float* st = ws + 16;
  const int masks[7] = {1, 2, 4, 5, 6, 3, 7};  // {1},{2},{3},{13},{23},{12},{123}
  float aCE[7] = {0, 0, 0, 0, 0, 0, 0};
  float aHt[7] = {0, 0, 0, 0, 0, 0, 0};
  for (int b = t; b < NROWS; b += 256) {
    float m[3], S[3], U[3], T[3], mo[3], SO[3];
    for (int k = 0; k < 3; k++) {
      const float* p = st + ((size_t)k * NROWS + b) * 6;
      m[k] = p[0]; S[k] = p[1]; U[k] = p[2]; T[k] = p[3]; mo[k] = p[4]; SO[k] = p[5];
    }
    for (int g = 0; g < 7; g++) {
      int msk = masks[g];
      float Md = -INFINITY, Mo = -INFINITY;
      for (int k = 0; k < 3; k++)
        if ((msk >> k) & 1) { Md = fmaxf(Md, m[k]); Mo = fmaxf(Mo, mo[k]); }
      float Zd = 0, Zo = 0, Tn = 0, Un = 0;
      for (int k = 0; k < 3; k++)
        if ((msk >> k) & 1) {
          float w = __expf(m[k] - Md);
          Zd += w * S[k]; Tn += w * T[k]; Un += w * U[k];
          Zo += __expf(mo[k] - Mo) * SO[k];
        }
      aCE[g] += Mo + __logf(Zo) - Tn / Zd;  // -sum t*logp
      aHt[g] += Md + __logf(Zd) - Un / Zd;  // -sum t*log t
    }
  }
  for (int g = 0; g < 7; g++) {
    atomicAdd(&red[g], aCE[g]);
    atomicAdd(&red[7 + g], aHt[g]);
  }
  __syncthreads();
  if (t == 0) {
    const float Bf = (float)NROWS;
    const int sz[7] = {1, 1, 1, 2, 2, 2, 3};
    float Ho[7];
    for (int g = 0; g < 7; g++) {
      float J = 1024.0f * (float)sz[g];
      Ho[g] = (1.0f - 1.0f / J) * (red[g] / Bf) + (1.0f / J) * (red[7 + g] / Bf);
    }
    float Hd1 = ws[4] / Bf, Hd2 = ws[5] / Bf, Hd3 = ws[6] / Bf;
    float Hd13 = ws[7] / Bf, Hd23 = ws[8] / Bf, Hd12 = ws[9] / Bf;
    float data_mu = ws[7] + ws[8] - ws[6] - ws[10];  // SUM over rows
    float H1 = Hd1 - Ho[0], H2 = Hd2 - Ho[1], H3 = Hd3 - Ho[2];
    float MI13 = (Ho[0] + Ho[2] - Ho[3]) - (Hd1 + Hd3 - Hd13);
    float MI23 = (Ho[1] + Ho[2] - Ho[4]) - (Hd2 + Hd3 - Hd23);
    float MI12 = (Ho[0] + Ho[1] - Ho[5]) - (Hd1 + Hd2 - Hd12);
    float CMI = (Ho[4] - Ho[2] + Ho[3] - Ho[6]) - data_mu;
    float mse = 0.5f * ws[3] / ((float)NROWS * 3072.0f);
    out[0] = 0.9f * mse + 0.1f * (H1 * H1 + H2 * H2 + H3 * H3 +
                                  MI13 * MI13 + MI23 * MI23 + MI12 * MI12 + CMI * CMI);
  }
}

extern "C" void kernel_launch(void* const* d_in, const int* in_sizes, int n_in,
                              void* d_out, int out_size, void* d_ws, size_t ws_size,
                              hipStream_t stream) {
  (void)in_sizes; (void)n_in; (void)out_size; (void)ws_size;
  const float* data = (const float*)d_in[0];
  const float* d1 = (const float*)d_in[1];
  const float* d2 = (const float*)d_in[2];
  const float* d3 = (const float*)d_in[3];
  const float* o1 = (const float*)d_in[4];
  const float* o2 = (const float*)d_in[5];
  const float* o3 = (const float*)d_in[6];
  const float* outp = (const float*)d_in[7];
  float* ws = (float*)d_ws;
  float* out = (float*)d_out;

  hipLaunchKernelGGL(init_kernel, dim3(1), dim3(64), 0, stream, ws);
  hipLaunchKernelGGL(stats_kernel, dim3(256, 3), dim3(256), 0, stream,
                     d1, d2, d3, o1, o2, o3, ws);
  hipLaunchKernelGGL(entropy_kernel, dim3(4096), dim3(256), 0, stream, d1, d2, d3, ws);
  hipLaunchKernelGGL(mse_kernel, dim3(2048), dim3(256), 0, stream, data, outp, ws);
  hipLaunchKernelGGL(final_kernel, dim3(1), dim3(256), 0, stream, ws, out);
}